// unit_gcn_59614146068723
// MI455X (gfx1250) — compile-verified
//
#include <hip/hip_runtime.h>

#define N_ 64
#define C_ 64
#define T_ 300
#define V_ 25
#define S_ 3
#define I_ 16
#define O_ 64
#define NTV_ (N_ * T_ * V_)
#define TCHUNK 20
#define NCHUNK (T_ / TCHUNK)

typedef float v2f __attribute__((ext_vector_type(2)));
typedef float v8f __attribute__((ext_vector_type(8)));
typedef unsigned int u32x4 __attribute__((ext_vector_type(4)));
typedef int i32x4 __attribute__((ext_vector_type(4)));
typedef int i32x8 __attribute__((ext_vector_type(8)));

#define WMMA_F32(a, b, c) \
    __builtin_amdgcn_wmma_f32_16x16x4_f32(false, (a), false, (b), (short)0, (c), false, false)

__device__ __forceinline__ v8f v8zero() {
    v8f z = {0.f, 0.f, 0.f, 0.f, 0.f, 0.f, 0.f, 0.f};
    return z;
}

// LDS byte offset of a __shared__ object (flat shared address: offset in low 32b).
__device__ __forceinline__ unsigned lds_off(const void* p) {
    return (unsigned)(unsigned long long)p;
}

// ---------------------------------------------------------------------------
// TDM: load `nrows` rows of `rowlen` dwords from global (row stride
// g_row_stride dwords) into LDS at pitch lds_row_stride dwords, using the
// descriptor-iteration mode (D# per CDNA5 ISA ch.8). One TENSORcnt per call.
// ---------------------------------------------------------------------------
__device__ __forceinline__ void tdm_load_rows(const float* g, unsigned lds_byte,
                                              int rowlen, int nrows,
                                              int g_row_stride, int lds_row_stride) {
    unsigned long long ga = (unsigned long long)g;
    u32x4 g0;
    g0.x = 1u;                                              // count=1, user mode
    g0.y = lds_byte;                                        // lds_addr (bytes)
    g0.z = (unsigned)ga;                                    // global_addr[31:0]
    g0.w = ((unsigned)(ga >> 32) & 0x01FFFFFFu) | (2u << 30);  // addr[56:32] | type=2
    i32x8 g1;
    g1[0] = (2 << 16) | (1 << 19);            // data_size=4B, iterate_enable=1
    g1[1] = (rowlen & 0xFFFF) << 16;          // tensor_dim0[15:0]
    g1[2] = ((rowlen >> 16) & 0xFFFF) | (1 << 16);  // tensor_dim0[31:16] | tensor_dim1=1
    g1[3] = (rowlen & 0xFFFF) << 16;          // tensor_dim1 hi=0 | tile_dim0=rowlen
    g1[4] = 1;                                // tile_dim1=1, tile_dim2=0
    g1[5] = g_row_stride;                     // tensor_dim0_stride[31:0]
    g1[6] = 0;
    g1[7] = 0;
    i32x4 g2;
    g2[0] = 0;
    g2[1] = lds_row_stride;                   // lds_addr_increment (elements)
    g2[2] = g_row_stride;                     // global_addr_increment (elements)
    g2[3] = (nrows - 1) << 16;                // iterate_count (0 => 1x)
    i32x4 g3;
    g3[0] = 0; g3[1] = 0; g3[2] = 0; g3[3] = 0;
    i32x8 g4;
    g4[0] = 0; g4[1] = 0; g4[2] = 0; g4[3] = 0;
    g4[4] = 0; g4[5] = 0; g4[6] = 0; g4[7] = 0;
    __builtin_amdgcn_tensor_load_to_lds(g0, g1, g2, g3, g4, 0);
}

#define WAIT_TENSORCNT_1() asm volatile("s_wait_tensorcnt 0x1" ::: "memory")
#define WAIT_DSCNT_0() asm volatile("s_wait_dscnt 0x0" ::: "memory")

// A-fragment 16x4 (MxK), row-major source with leading dim ld (ISA 7.12.2).
__device__ __forceinline__ v2f frag_a(const float* src, int ld, int lane) {
    int m = lane & 15;
    int k = (lane >> 4) << 1;
    v2f a;
    a.x = src[m * ld + k];
    a.y = src[m * ld + k + 1];
    return a;
}

// A-fragment where logical A[m][k] = src[k*ld + m] (transposed source).
__device__ __forceinline__ v2f frag_at(const float* src, int ld, int lane) {
    int m = lane & 15;
    int k = (lane >> 4) << 1;
    v2f a;
    a.x = src[k * ld + m];
    a.y = src[(k + 1) * ld + m];
    return a;
}

// B-fragment 4x16 (KxN), row-major source.
__device__ __forceinline__ v2f frag_b(const float* src, int ld, int lane) {
    int n = lane & 15;
    int k = (lane >> 4) << 1;
    v2f b;
    b.x = src[k * ld + n];
    b.y = src[(k + 1) * ld + n];
    return b;
}

// ---------------------------------------------------------------------------
// Kernel 1: attention scores. One wave per (n,s).
// att[n,s,v,w] = (1/(I*T)) * sum_t (Wa_s Xt + ba)^T (Wb_s Xt + bb)
// x tiles stream through LDS via double-buffered TDM loads.
// ---------------------------------------------------------------------------
__global__ __launch_bounds__(32) void k_att(const float* __restrict__ x,
                                            const float* __restrict__ Wa,
                                            const float* __restrict__ ba,
                                            const float* __restrict__ Wb,
                                            const float* __restrict__ bb,
                                            float* __restrict__ att) {
    __shared__ float lds_wa[I_ * C_];
    __shared__ float lds_wb[I_ * C_];
    __shared__ float lds_x[2][C_ * 32];
    __shared__ float lds_fa[I_ * 32];
    __shared__ float lds_fb[I_ * 32];
    __shared__ float lds_ba[I_];
    __shared__ float lds_bb[I_];

    const int n = blockIdx.x / S_;
    const int s = blockIdx.x % S_;
    const int lane = threadIdx.x;
    const int col = lane & 15;
    const int roff = (lane >> 4) * 8;

    // zero x buffers once (TDM only writes cols 0..24; pad cols stay 0)
    for (int i = lane; i < C_ * 32; i += 32) {
        lds_x[0][i] = 0.f;
        lds_x[1][i] = 0.f;
    }
    if (lane < I_) {
        lds_ba[lane] = ba[s * I_ + lane];
        lds_bb[lane] = bb[s * I_ + lane];
    }
    WAIT_DSCNT_0();  // DS zero-stores complete before TDM writes same lines

    const float* xn = x + (size_t)n * C_ * T_ * V_;
    tdm_load_rows(Wa + s * I_ * C_, lds_off(lds_wa), I_ * C_, 1, 0, 0);
    tdm_load_rows(Wb + s * I_ * C_, lds_off(lds_wb), I_ * C_, 1, 0, 0);
    tdm_load_rows(xn, lds_off(lds_x[0]), V_, C_, T_ * V_, 32);  // tile t=0
    __syncthreads();

    v8f c00 = v8zero(), c01 = v8zero(), c10 = v8zero(), c11 = v8zero();

    for (int t = 0; t < T_; ++t) {
        // prefetch next tile into the other buffer (dummy reload of t=0 at end)
        int tn = (t + 1 < T_) ? t + 1 : 0;
        tdm_load_rows(xn + tn * V_, lds_off(lds_x[(t + 1) & 1]), V_, C_, T_ * V_, 32);
        WAIT_TENSORCNT_1();  // all older transfers (incl. tile t) complete
        __syncthreads();
        const float* xb = lds_x[t & 1];

        // Fa = Wa_s @ Xt, Fb = Wb_s @ Xt : M=16, N=32(2 tiles), K=64
        v8f fa0 = v8zero(), fa1 = v8zero(), fb0 = v8zero(), fb1 = v8zero();
#pragma unroll
        for (int k = 0; k < C_; k += 4) {
            v2f b0 = frag_b(xb + k * 32, 32, lane);
            v2f b1 = frag_b(xb + k * 32 + 16, 32, lane);
            v2f aa = frag_a(lds_wa + k, C_, lane);
            fa0 = WMMA_F32(aa, b0, fa0);
            fa1 = WMMA_F32(aa, b1, fa1);
            v2f ab = frag_a(lds_wb + k, C_, lane);
            fb0 = WMMA_F32(ab, b0, fb0);
            fb1 = WMMA_F32(ab, b1, fb1);
        }
        // add bias, stage in LDS for re-striping
#pragma unroll
        for (int j = 0; j < 8; ++j) {
            int r = roff + j;
            lds_fa[r * 32 + col] = fa0[j] + lds_ba[r];
            lds_fa[r * 32 + col + 16] = fa1[j] + lds_ba[r];
            lds_fb[r * 32 + col] = fb0[j] + lds_bb[r];
            lds_fb[r * 32 + col + 16] = fb1[j] + lds_bb[r];
        }
        __syncthreads();

        // att += Fa^T @ Fb : M=32(2), N=32(2), K=16
#pragma unroll
        for (int k = 0; k < I_; k += 4) {
            v2f a0 = frag_at(lds_fa + k * 32, 32, lane);
            v2f a1 = frag_at(lds_fa + k * 32 + 16, 32, lane);
            v2f b0 = frag_b(lds_fb + k * 32, 32, lane);
            v2f b1 = frag_b(lds_fb + k * 32 + 16, 32, lane);
            c00 = WMMA_F32(a0, b0, c00);
            c01 = WMMA_F32(a0, b1, c01);
            c10 = WMMA_F32(a1, b0, c10);
            c11 = WMMA_F32(a1, b1, c11);
        }
    }

    const float inv = 1.0f / (float)(I_ * T_);
    float* dst = att + (n * S_ + s) * V_ * V_;
#pragma unroll
    for (int j = 0; j < 8; ++j) {
        int v0 = roff + j;
        int v1 = v0 + 16;
        if (v0 < V_ && col < V_) dst[v0 * V_ + col] = c00[j] * inv;
        if (v0 < V_ && col + 16 < V_) dst[v0 * V_ + col + 16] = c01[j] * inv;
        if (v1 < V_ && col < V_) dst[v1 * V_ + col] = c10[j] * inv;
        if (v1 < V_ && col + 16 < V_) dst[v1 * V_ + col + 16] = c11[j] * inv;
    }
}

// ---------------------------------------------------------------------------
// Kernel 2: softmax over v (axis -2) per (n,s,w) column, then + (A + PA).
// ---------------------------------------------------------------------------
__global__ __launch_bounds__(32) void k_softmax(float* __restrict__ att,
                                                const float* __restrict__ A,
                                                const float* __restrict__ PA) {
    const int n = blockIdx.x / S_;
    const int s = blockIdx.x % S_;
    const int w = threadIdx.x;
    if (w >= V_) return;
    float* base = att + (n * S_ + s) * V_ * V_;
    const float* Ab = A + s * V_ * V_;
    const float* Pb = PA + s * V_ * V_;
    float m = -1e30f;
    for (int v = 0; v < V_; ++v) m = fmaxf(m, base[v * V_ + w]);
    float sum = 0.f;
    for (int v = 0; v < V_; ++v) sum += __expf(base[v * V_ + w] - m);
    float rs = 1.0f / sum;
    for (int v = 0; v < V_; ++v) {
        float e = __expf(base[v * V_ + w] - m) * rs;
        base[v * V_ + w] = e + Ab[v * V_ + w] + Pb[v * V_ + w];
    }
}

// ---------------------------------------------------------------------------
// Kernel 3: message passing + output conv. One wave per (n, t-chunk).
// Y[n,:,t,:] = sum_s Wd_s @ (Xt @ att[n,s]) + sum_s bd_s  -> d_out (pre-BN)
// Wd / att staged via TDM; x tiles double-buffered via TDM.
// ---------------------------------------------------------------------------
__global__ __launch_bounds__(32) void k_msg(const float* __restrict__ x,
                                            const float* __restrict__ att,
                                            const float* __restrict__ Wd,
                                            const float* __restrict__ bd,
                                            float* __restrict__ y) {
    __shared__ float lds_wd[S_ * O_ * C_];   // 48 KB
    __shared__ float lds_att[S_ * 32 * 32];  // 12 KB (zero padded)
    __shared__ float lds_x[2][C_ * 32];      // 16 KB double buffer
    __shared__ float lds_z[C_ * 32];         // 8 KB
    __shared__ float lds_bd[O_];

    const int n = blockIdx.x / NCHUNK;
    const int t0 = (blockIdx.x % NCHUNK) * TCHUNK;
    const int lane = threadIdx.x;
    const int col = lane & 15;
    const int roff = (lane >> 4) * 8;

    for (int i = lane; i < S_ * 32 * 32; i += 32) lds_att[i] = 0.f;
    for (int i = lane; i < C_ * 32; i += 32) {
        lds_x[0][i] = 0.f;
        lds_x[1][i] = 0.f;
    }
    for (int o = lane; o < O_; o += 32) lds_bd[o] = bd[o] + bd[O_ + o] + bd[2 * O_ + o];
    WAIT_DSCNT_0();  // zero-stores complete before TDM writes

    tdm_load_rows(Wd, lds_off(lds_wd), S_ * O_ * C_, 1, 0, 0);
    for (int s = 0; s < S_; ++s)
        tdm_load_rows(att + (n * S_ + s) * V_ * V_, lds_off(lds_att) + s * 4096, V_, V_, V_, 32);
    const float* xn = x + (size_t)n * C_ * T_ * V_;
    tdm_load_rows(xn + t0 * V_, lds_off(lds_x[0]), V_, C_, T_ * V_, 32);
    __syncthreads();

    for (int t = t0; t < t0 + TCHUNK; ++t) {
        int p = (t - t0) & 1;
        int tn = (t + 1 < t0 + TCHUNK) ? t + 1 : t0;  // dummy reload at end
        tdm_load_rows(xn + tn * V_, lds_off(lds_x[p ^ 1]), V_, C_, T_ * V_, 32);
        WAIT_TENSORCNT_1();  // all older transfers (incl. tile t) complete
        __syncthreads();
        const float* xb = lds_x[p];

        v8f yf[4][2];
#pragma unroll
        for (int mi = 0; mi < 4; ++mi) {
            yf[mi][0] = v8zero();
            yf[mi][1] = v8zero();
        }

        for (int s = 0; s < S_; ++s) {
            // Z = Xt @ att_s : M=64(4), N=32(2), K=32 (padded)
            v8f zf[4][2];
#pragma unroll
            for (int mi = 0; mi < 4; ++mi) {
                zf[mi][0] = v8zero();
                zf[mi][1] = v8zero();
            }
#pragma unroll
            for (int k = 0; k < 32; k += 4) {
                v2f b0 = frag_b(lds_att + s * 1024 + k * 32, 32, lane);
                v2f b1 = frag_b(lds_att + s * 1024 + k * 32 + 16, 32, lane);
#pragma unroll
                for (int mi = 0; mi < 4; ++mi) {
                    v2f a = frag_a(xb + mi * 16 * 32 + k, 32, lane);
                    zf[mi][0] = WMMA_F32(a, b0, zf[mi][0]);
                    zf[mi][1] = WMMA_F32(a, b1, zf[mi][1]);
                }
            }
            __syncthreads();  // previous s reads of lds_z finished
#pragma unroll
            for (int mi = 0; mi < 4; ++mi) {
#pragma unroll
                for (int j = 0; j < 8; ++j) {
                    int r = mi * 16 + roff + j;
                    lds_z[r * 32 + col] = zf[mi][0][j];
                    lds_z[r * 32 + col + 16] = zf[mi][1][j];
                }
            }
            __syncthreads();

            // Y += Wd_s @ Z : M=64(4), N=32(2), K=64
#pragma unroll
            for (int k = 0; k < C_; k += 4) {
                v2f b0 = frag_b(lds_z + k * 32, 32, lane);
                v2f b1 = frag_b(lds_z + k * 32 + 16, 32, lane);
#pragma unroll
                for (int mi = 0; mi < 4; ++mi) {
                    v2f a = frag_a(lds_wd + s * O_ * C_ + mi * 16 * C_ + k, C_, lane);
                    yf[mi][0] = WMMA_F32(a, b0, yf[mi][0]);
                    yf[mi][1] = WMMA_F32(a, b1, yf[mi][1]);
                }
            }
        }

#pragma unroll
        for (int mi = 0; mi < 4; ++mi) {
#pragma unroll
            for (int j = 0; j < 8; ++j) {
                int o = mi * 16 + roff + j;
                float bias = lds_bd[o];
                float* dst = y + ((n * O_ + o) * T_ + t) * V_;
                if (col < V_) dst[col] = yf[mi][0][j] + bias;
                if (col + 16 < V_) dst[col + 16] = yf[mi][1][j] + bias;
            }
        }
    }
}

// ---------------------------------------------------------------------------
// Kernel 4: per-channel sum / sumsq over (N,T,V). Block per (n,o).
// ---------------------------------------------------------------------------
__global__ __launch_bounds__(256) void k_stats(const float* __restrict__ y,
                                               float* __restrict__ sums,
                                               float* __restrict__ sumsq) {
    __shared__ float s1[256];
    __shared__ float s2[256];
    const int no = blockIdx.x;
    const int o = no & (O_ - 1);
    const float* base = y + (size_t)no * (T_ * V_);
    float a = 0.f, b = 0.f;
    for (int i = threadIdx.x; i < T_ * V_; i += 256) {
        float v = base[i];
        a += v;
        b += v * v;
    }
    s1[threadIdx.x] = a;
    s2[threadIdx.x] = b;
    __syncthreads();
    for (int st = 128; st > 0; st >>= 1) {
        if (threadIdx.x < st) {
            s1[threadIdx.x] += s1[threadIdx.x + st];
            s2[threadIdx.x] += s2[threadIdx.x + st];
        }
        __syncthreads();
    }
    if (threadIdx.x == 0) {
        atomicAdd(&sums[o], s1[0]);
        atomicAdd(&sumsq[o], s2[0]);
    }
}

// ---------------------------------------------------------------------------
// Kernel 5: fold batch stats + gamma/beta into scale/shift.
// ---------------------------------------------------------------------------
__global__ void k_bnfinal(const float* __restrict__ sums, const float* __restrict__ sumsq,
                          const float* __restrict__ gamma, const float* __restrict__ beta,
                          float* __restrict__ scale, float* __restrict__ shift) {
    int o = threadIdx.x;
    if (o < O_) {
        float inv = 1.0f / (float)NTV_;
        float mean = sums[o] * inv;
        float var = sumsq[o] * inv - mean * mean;
        float sc = gamma[o] * rsqrtf(var + 1e-5f);
        scale[o] = sc;
        shift[o] = beta[o] - mean * sc;
    }
}

// ---------------------------------------------------------------------------
// Kernel 6: in-place BN + residual + relu.
// ---------------------------------------------------------------------------
__global__ __launch_bounds__(256) void k_apply(float* __restrict__ y, const float* __restrict__ x,
                                               const float* __restrict__ scale,
                                               const float* __restrict__ shift) {
    int i = blockIdx.x * 256 + threadIdx.x;
    if (i < N_ * O_ * T_ * V_) {
        int o = (i / (T_ * V_)) & (O_ - 1);
        float v = y[i] * scale[o] + shift[o] + x[i];
        y[i] = fmaxf(v, 0.f);
    }
}

extern "C" void kernel_launch(void* const* d_in, const int* in_sizes, int n_in,
                              void* d_out, int out_size, void* d_ws, size_t ws_size,
                              hipStream_t stream) {
    const float* x = (const float*)d_in[0];
    const float* A = (const float*)d_in[1];
    const float* PA = (const float*)d_in[2];
    const float* Wa = (const float*)d_in[3];
    const float* ba = (const float*)d_in[4];
    const float* Wb = (const float*)d_in[5];
    const float* bb = (const float*)d_in[6];
    const float* Wd = (const float*)d_in[7];
    const float* bd = (const float*)d_in[8];
    const float* gamma = (const float*)d_in[9];
    const float* beta = (const float*)d_in[10];
    float* y = (float*)d_out;

    float* ws = (float*)d_ws;
    float* att = ws;                 // N*S*V*V = 120000 floats
    float* sums = ws + 120000;       // 64
    float* sumsq = ws + 120064;      // 64
    float* scale = ws + 120128;      // 64
    float* shift = ws + 120192;      // 64

    (void)hipMemsetAsync(sums, 0, 2 * O_ * sizeof(float), stream);

    k_att<<<dim3(N_ * S_), dim3(32), 0, stream>>>(x, Wa, ba, Wb, bb, att);
    k_softmax<<<dim3(N_ * S_), dim3(32), 0, stream>>>(att, A, PA);
    k_msg<<<dim3(N_ * NCHUNK), dim3(32), 0, stream>>>(x, att, Wd, bd, y);
    k_stats<<<dim3(N_ * O_), dim3(256), 0, stream>>>(y, sums, sumsq);
    k_bnfinal<<<dim3(1), dim3(64), 0, stream>>>(sums, sumsq, gamma, beta, scale, shift);
    int total = N_ * O_ * T_ * V_;
    k_apply<<<dim3((total + 255) / 256), dim3(256), 0, stream>>>(y, x, scale, shift);
}